// HarmonicGenerator_2233382993940
// MI455X (gfx1250) — compile-verified
//
#include <hip/hip_runtime.h>
#include <math.h>

// ---------------- problem constants ----------------
#define SRf   48000.0f
#define SEGi  960
#define NHh   8
#define NBn   16
#define CCc   256
#define LFl   200
#define LWw   (LFl * SEGi)          // 192000
#define SPT   3                     // samples per thread in scan kernels
#define SBLK  256                   // threads per scan block
#define SPB   (SPT * SBLK)          // 768 samples per block
#define NBLK  (LWw / SPB)           // 250 blocks per batch

// ---------------- workspace layout (float offsets, all 8-float aligned) ----
#define WS_W     0                  // 16 x 256 combined weight matrix
#define WS_SCAL  4096               // [0]=A=sum(scale*oct_w), [1]=Bc=sum(shift*oct_w)
#define WS_MU    4104               // N*LF
#define WS_ISIG  7304               // N*LF
#define WS_F0    10504              // N*LF
#define WS_BSUM  13704              // N*NBLK
#define WS_BOFF  17704              // N*NBLK
#define WS_MAGT  21704              // N*LF*8  (magT[n][l][h], h contiguous)
// total 47304 floats = 189,216 bytes

typedef __attribute__((ext_vector_type(2))) float v2f;
typedef __attribute__((ext_vector_type(8))) float v8f;

// ---------------------------------------------------------------------------
// Kernel 0: build combined weight matrix (rows 0..7 = mag_w, row 8 = scale*oct_w,
// rows 9..15 = 0) and the two fold-in scalars A, Bc.
// ---------------------------------------------------------------------------
__global__ void __launch_bounds__(256) hg_prep_kernel(
    const float* __restrict__ mag_w, const float* __restrict__ cn_scale,
    const float* __restrict__ cn_shift, const float* __restrict__ oct_w,
    float* __restrict__ ws)
{
    int c = threadIdx.x;                       // 0..255
    float ow = oct_w[c];
    float sc = cn_scale[c];
    float sh = cn_shift[c];
    float* W = ws + WS_W;
#pragma unroll
    for (int h = 0; h < NHh; ++h) W[h * CCc + c] = mag_w[h * CCc + c];
    W[8 * CCc + c] = sc * ow;
#pragma unroll
    for (int h = 9; h < 16; ++h) W[h * CCc + c] = 0.0f;

    __shared__ float sA[256], sB[256];
    sA[c] = sc * ow;
    sB[c] = sh * ow;
    __syncthreads();
    for (int off = 128; off > 0; off >>= 1) {
        if (c < off) { sA[c] += sA[c + off]; sB[c] += sB[c + off]; }
        __syncthreads();
    }
    if (c == 0) { ws[WS_SCAL] = sA[0]; ws[WS_SCAL + 1] = sB[0]; }
}

// ---------------------------------------------------------------------------
// Kernel 1: per-(n,l) channel statistics. One thread per (n,l); adjacent
// threads share n, consecutive l -> coalesced loads for each c.
// ---------------------------------------------------------------------------
__global__ void __launch_bounds__(256) hg_stats_kernel(
    const float* __restrict__ x, float* __restrict__ ws)
{
    int tid = blockIdx.x * blockDim.x + threadIdx.x;
    if (tid >= NBn * LFl) return;
    int n = tid / LFl, l = tid % LFl;
    const float* xp = x + (size_t)n * CCc * LFl + l;
    float s = 0.0f, ss = 0.0f;
#pragma unroll 8
    for (int c = 0; c < CCc; ++c) {
        float v = xp[(size_t)c * LFl];
        s += v; ss += v * v;
    }
    float mu  = s * (1.0f / CCc);
    float var = fmaxf((ss - s * mu) * (1.0f / (CCc - 1)), 0.0f); // unbiased
    float sig = sqrtf(var) + 1e-4f;
    ws[WS_MU   + tid] = mu;
    ws[WS_ISIG + tid] = 1.0f / sig;
}

// ---------------------------------------------------------------------------
// Kernel 2: WMMA projection. One wave per (l-tile, n).
//   D(16x16) = X_tile(16 l x 256 c) * Wcomb^T(256 c x 16 out), K-steps of 4
//   using V_WMMA_F32_16X16X4_F32 (native f32 path; data is f32).
// A layout (16x4 f32): lanes 0-15 m=lane, v0=K0,v1=K1; lanes 16-31 v0=K2,v1=K3.
// B layout (4x16 f32): lanes 0-15 nOut=lane, v0=K0,v1=K1; lanes 16-31 K2/K3.
// C/D: lane%16 = N(out), VGPR r -> M = r + 8*(lane/16).
// Epilogue: out columns 0..7 -> mag (exp/min), column 8 -> f0 via folded norm.
// ---------------------------------------------------------------------------
__global__ void __launch_bounds__(32) hg_proj_wmma_kernel(
    const float* __restrict__ x, const float* __restrict__ mag_b,
    const float* __restrict__ oct_b, float* __restrict__ ws)
{
    __shared__ float lw[16 * CCc];             // 16 KB weight stage
    int lane = threadIdx.x;
    const float* Wc = ws + WS_W;
    for (int i = lane; i < 16 * CCc; i += 32) lw[i] = Wc[i];
    __syncthreads();

    int n  = blockIdx.y;
    int l0 = blockIdx.x * 16;
    int half = lane >> 4;                      // 0/1 -> K sub-pair
    int idx  = lane & 15;                      // A: row m, B: out column
    int lA = l0 + idx; if (lA > LFl - 1) lA = LFl - 1;   // clamp (discard later)
    const float* xbase = x + (size_t)n * CCc * LFl + lA;

    v8f acc = {};
    for (int k = 0; k < 64; ++k) {             // K = 256 in steps of 4
        int c = 4 * k + 2 * half;
        v2f a, b;
        a.x = xbase[(size_t)c * LFl];
        a.y = xbase[(size_t)(c + 1) * LFl];
        b.x = lw[idx * CCc + c];
        b.y = lw[idx * CCc + c + 1];
        acc = __builtin_amdgcn_wmma_f32_16x16x4_f32(
                  false, a, false, b, (short)0, acc, false, false);
    }

    float A  = ws[WS_SCAL];
    float Bc = ws[WS_SCAL + 1];
    float ob = oct_b[0];
    float mb = (idx < NHh) ? mag_b[idx] : 0.0f;
#pragma unroll
    for (int r = 0; r < 8; ++r) {
        int l = l0 + r + 8 * half;             // D row -> l position
        if (l >= LFl) continue;
        float v = acc[r];
        if (idx < NHh) {
            ws[WS_MAGT + ((size_t)n * LFl + l) * NHh + idx] =
                __expf(fminf(v + mb, 6.0f));
        } else if (idx == 8) {
            float mu = ws[WS_MU   + n * LFl + l];
            float is = ws[WS_ISIG + n * LFl + l];
            float oct = (v - mu * A) * is + Bc + ob;
            float f0  = fminf(fmaxf(440.0f * exp2f(oct), 20.0f), 16000.0f);
            ws[WS_F0 + n * LFl + l] = f0;
        }
    }
}

// ---------------------------------------------------------------------------
// interp helper: F[j] = linear upsample of f0 row (align_corners=False style)
// ---------------------------------------------------------------------------
__device__ __forceinline__ float hg_interpF(const float* __restrict__ row, int j,
                                            int& i0, int& i1, float& w)
{
    float src = ((float)j + 0.5f) * (1.0f / (float)SEGi) - 0.5f;
    src = fminf(fmaxf(src, 0.0f), (float)(LFl - 1));
    i0 = (int)src;                              // floor (src >= 0)
    i1 = (i0 + 1 < LFl) ? i0 + 1 : LFl - 1;
    w  = src - (float)i0;
    return row[i0] * (1.0f - w) + row[i1] * w;
}

// ---------------------------------------------------------------------------
// Kernel 3: per-block partial sums of F (in F units, /SR applied at the end).
// ---------------------------------------------------------------------------
__global__ void __launch_bounds__(SBLK) hg_blocksum_kernel(float* __restrict__ ws)
{
    int n = blockIdx.y, b = blockIdx.x, t = threadIdx.x;
    const float* row = ws + WS_F0 + n * LFl;
    int jbase = b * SPB + t * SPT;
    float s = 0.0f;
#pragma unroll
    for (int i = 0; i < SPT; ++i) { int a0, a1; float w;
        s += hg_interpF(row, jbase + i, a0, a1, w); }
    __shared__ float red[SBLK];
    red[t] = s; __syncthreads();
    for (int off = SBLK / 2; off > 0; off >>= 1) {
        if (t < off) red[t] += red[t + off];
        __syncthreads();
    }
    if (t == 0) ws[WS_BSUM + n * NBLK + b] = red[0];
}

// ---------------------------------------------------------------------------
// Kernel 4: tiny serial exclusive scan of the 250 block sums per batch.
// ---------------------------------------------------------------------------
__global__ void __launch_bounds__(32) hg_blockscan_kernel(float* __restrict__ ws)
{
    int t = threadIdx.x;
    if (t < NBn) {
        float run = 0.0f;
        for (int b = 0; b < NBLK; ++b) {
            ws[WS_BOFF + t * NBLK + b] = run;
            run += ws[WS_BSUM + t * NBLK + b];
        }
    }
}

// ---------------------------------------------------------------------------
// Kernel 5: fused intra-block scan + harmonic synthesis.
//   T[j] = (blockOff + exclusive + local prefix) / SR
//   wave[j] = (1/8) * sum_h sin(2*pi*h*T[j]) * lerp(magT[n][i0][h], magT[n][i1][h])
// ---------------------------------------------------------------------------
__global__ void __launch_bounds__(SBLK) hg_synth_kernel(
    float* __restrict__ ws, float* __restrict__ out)
{
    int n = blockIdx.y, b = blockIdx.x, t = threadIdx.x;
    const float* row  = ws + WS_F0 + n * LFl;
    const float* magT = ws + WS_MAGT + (size_t)n * LFl * NHh;
    int jbase = b * SPB + t * SPT;

    int   I0[SPT], I1[SPT];
    float Wt[SPT], pref[SPT];
    float run = 0.0f;
#pragma unroll
    for (int i = 0; i < SPT; ++i) {
        run += hg_interpF(row, jbase + i, I0[i], I1[i], Wt[i]);
        pref[i] = run;                          // inclusive within thread
    }

    __shared__ float s[SBLK];                   // Hillis-Steele inclusive scan
    s[t] = run; __syncthreads();
    for (int off = 1; off < SBLK; off <<= 1) {
        float v = (t >= off) ? s[t - off] : 0.0f;
        __syncthreads();
        s[t] += v;
        __syncthreads();
    }
    float base = ws[WS_BOFF + n * NBLK + b] + (s[t] - run);  // exclusive

#pragma unroll
    for (int i = 0; i < SPT; ++i) {
        float T = (base + pref[i]) * (1.0f / SRf);
        const float* m0 = magT + I0[i] * NHh;
        const float* m1 = magT + I1[i] * NHh;
        float w = Wt[i];
        float acc = 0.0f;
#pragma unroll
        for (int h = 0; h < NHh; ++h) {
            float m  = m0[h] + (m1[h] - m0[h]) * w;
            float ph = T * (float)(h + 1);
            float fr = ph - floorf(ph);         // sin(2*pi*x) periodic reduce
            acc += __sinf(6.283185307179586f * fr) * m;
        }
        out[(size_t)n * LWw + jbase + i] = acc * (1.0f / NHh);
    }
}

// ---------------------------------------------------------------------------
extern "C" void kernel_launch(void* const* d_in, const int* in_sizes, int n_in,
                              void* d_out, int out_size, void* d_ws, size_t ws_size,
                              hipStream_t stream)
{
    (void)in_sizes; (void)n_in; (void)out_size; (void)ws_size;
    const float* x        = (const float*)d_in[0];   // (16,256,200)
    const float* mag_w    = (const float*)d_in[1];   // (8,256)
    const float* mag_b    = (const float*)d_in[2];   // (8,)
    const float* cn_scale = (const float*)d_in[3];   // (1,256,1)
    const float* cn_shift = (const float*)d_in[4];   // (1,256,1)
    const float* oct_w    = (const float*)d_in[5];   // (1,256)
    const float* oct_b    = (const float*)d_in[6];   // (1,)
    float* out = (float*)d_out;                      // (16,1,192000)
    float* ws  = (float*)d_ws;                       // ~189 KB used

    hg_prep_kernel<<<1, 256, 0, stream>>>(mag_w, cn_scale, cn_shift, oct_w, ws);

    int statThreads = NBn * LFl;                     // 3200
    hg_stats_kernel<<<(statThreads + 255) / 256, 256, 0, stream>>>(x, ws);

    dim3 gProj((LFl + 15) / 16, NBn);                // 13 x 16 waves
    hg_proj_wmma_kernel<<<gProj, 32, 0, stream>>>(x, mag_b, oct_b, ws);

    dim3 gScan(NBLK, NBn);                           // 250 x 16
    hg_blocksum_kernel<<<gScan, SBLK, 0, stream>>>(ws);
    hg_blockscan_kernel<<<1, 32, 0, stream>>>(ws);
    hg_synth_kernel<<<gScan, SBLK, 0, stream>>>(ws, out);
}